// VariationalDist_76261439308501
// MI455X (gfx1250) — compile-verified
//
#include <hip/hip_runtime.h>
#include <hip/hip_bf16.h>

// ---------------------------------------------------------------------------
// VariationalDist graph propagation for MI455X (gfx1250, wave32).
// Bandwidth/atomic bound: no dense matmul structure -> no WMMA.
// gfx1250-specific paths: async global->LDS edge staging (ASYNCcnt, 4 deep),
// global_prefetch, f32 global atomics, 128-bit node-major gathers.
// ---------------------------------------------------------------------------

#ifndef __has_builtin
#define __has_builtin(x) 0
#endif

typedef int vd_v2i __attribute__((ext_vector_type(2)));

#if __has_builtin(__builtin_amdgcn_global_load_async_to_lds_b64)
#define VD_ASYNC_LDS 1
typedef __attribute__((address_space(1))) vd_v2i* vd_g64p;
typedef __attribute__((address_space(3))) vd_v2i* vd_l64p;
#else
#define VD_ASYNC_LDS 0
#endif

__device__ __forceinline__ void vd_wait_async0() {
#if __has_builtin(__builtin_amdgcn_s_wait_asynccnt)
  __builtin_amdgcn_s_wait_asynccnt(0);
#else
  asm volatile("s_wait_asynccnt 0" ::: "memory");
#endif
}

__device__ __forceinline__ float vd_softplus(float v) {
  // numerically stable log(1+exp(v))
  return fmaxf(v, 0.0f) + log1pf(expf(-fabsf(v)));
}

// ---------------------------------------------------------------------------
// Kernel 1: edge prep. int32 edge_index [2,E] -> packed (src,dst) pairs,
// and degree(src) via f32 atomics.
// ---------------------------------------------------------------------------
__global__ void vd_prep_edges(const int* __restrict__ ei, int2* __restrict__ pairs,
                              float* __restrict__ deg, long long E) {
  long long e = (long long)blockIdx.x * blockDim.x + threadIdx.x;
  if (e >= E) return;
  int s = ei[e];
  int d = ei[E + e];
  pairs[e] = make_int2(s, d);
  atomicAdd(&deg[s], 1.0f);
}

// ---------------------------------------------------------------------------
// Kernel 2: x[n][s] = softplus(diag[n]) * sample[s][n]   (transpose to [N][S])
// ---------------------------------------------------------------------------
__global__ void vd_init_x(const float* __restrict__ sample, const float* __restrict__ diag,
                          float* __restrict__ x, int n_nodes, int S) {
  int n = blockIdx.x * blockDim.x + threadIdx.x;
  if (n >= n_nodes) return;
  float sp = vd_softplus(diag[n]);
  for (int s = 0; s < S; ++s)
    x[(size_t)n * S + s] = sp * sample[(size_t)s * n_nodes + n];
}

// ---------------------------------------------------------------------------
// Kernel 3: unweighted scatter  agg[dst][:] += x[src][:]   (S == 16 fast path)
// Each thread owns VD_ILP edges per chunk; edge pairs are staged into LDS via
// gfx1250 async global->LDS b64 loads (all issued, then one asynccnt wait).
// ---------------------------------------------------------------------------
#define VD_B   256
#define VD_ILP 4

__global__ void __launch_bounds__(VD_B)
vd_scatter16(const int2* __restrict__ pairs, const float* __restrict__ x,
             float* __restrict__ agg, long long E) {
  __shared__ vd_v2i tile[VD_B * VD_ILP];
  const long long base = (long long)blockIdx.x * (VD_B * VD_ILP) + threadIdx.x;

  // prefetch the edge stream one chunk-group ahead (global_prefetch_b8)
  if (base + VD_B * VD_ILP * 8 < E)
    __builtin_prefetch(pairs + base + VD_B * VD_ILP * 8, 0, 0);

  // stage VD_ILP coalesced edge pairs per thread into LDS
#pragma unroll
  for (int i = 0; i < VD_ILP; ++i) {
    long long e = base + (long long)i * VD_B;
    if (e < E) {
#if VD_ASYNC_LDS
      __builtin_amdgcn_global_load_async_to_lds_b64(
          (vd_g64p)(pairs + e), (vd_l64p)&tile[i * VD_B + threadIdx.x], 0, 0);
#else
      tile[i * VD_B + threadIdx.x] = *(const vd_v2i*)(pairs + e);
#endif
    }
  }
#if VD_ASYNC_LDS
  vd_wait_async0();
#endif

#pragma unroll
  for (int i = 0; i < VD_ILP; ++i) {
    long long e = base + (long long)i * VD_B;
    if (e >= E) continue;
    vd_v2i p = tile[i * VD_B + threadIdx.x];
    int srcn = p.x;
    int dstn = p.y;

    const float4* xs = (const float4*)(x + (size_t)srcn * 16);
    float4 a0 = xs[0], a1 = xs[1], a2 = xs[2], a3 = xs[3];
    float* ag = agg + (size_t)dstn * 16;
    atomicAdd(ag + 0,  a0.x); atomicAdd(ag + 1,  a0.y);
    atomicAdd(ag + 2,  a0.z); atomicAdd(ag + 3,  a0.w);
    atomicAdd(ag + 4,  a1.x); atomicAdd(ag + 5,  a1.y);
    atomicAdd(ag + 6,  a1.z); atomicAdd(ag + 7,  a1.w);
    atomicAdd(ag + 8,  a2.x); atomicAdd(ag + 9,  a2.y);
    atomicAdd(ag + 10, a2.z); atomicAdd(ag + 11, a2.w);
    atomicAdd(ag + 12, a3.x); atomicAdd(ag + 13, a3.y);
    atomicAdd(ag + 14, a3.z); atomicAdd(ag + 15, a3.w);
  }
}

// generic S fallback (not used for the reference shapes)
__global__ void vd_scatter_gen(const int2* __restrict__ pairs, const float* __restrict__ x,
                               float* __restrict__ agg, long long E, int S) {
  long long e = (long long)blockIdx.x * blockDim.x + threadIdx.x;
  if (e >= E) return;
  int2 p = pairs[e];
  const float* xs = x + (size_t)p.x * S;
  float* ag = agg + (size_t)p.y * S;
  for (int s = 0; s < S; ++s) atomicAdd(ag + s, xs[s]);
}

// ---------------------------------------------------------------------------
// Kernel 4: per-node combine.
//   gamma = sigmoid(gp[l]); sw = exp(a1[l]); nw = sw*tanh(a2[l])
//   x[n][:] = sw*exp(gamma*log_deg[n]) * x[n][:]
//           + nw*exp((gamma-1)*log_deg[n]) * agg[n][:]
// ---------------------------------------------------------------------------
__global__ void vd_combine(float* __restrict__ x, const float* __restrict__ agg,
                           const float* __restrict__ deg,
                           const float* __restrict__ a1, const float* __restrict__ a2,
                           const float* __restrict__ gp, int l, int n_nodes, int S) {
  int n = blockIdx.x * blockDim.x + threadIdx.x;
  if (n >= n_nodes) return;
  float gamma = 1.0f / (1.0f + expf(-gp[l]));
  float sw = expf(a1[l]);
  float nw = sw * tanhf(a2[l]);
  float ld = logf(deg[n]);
  float ca = sw * expf(gamma * ld);
  float cb = nw * expf((gamma - 1.0f) * ld);
  float* xp = x + (size_t)n * S;
  const float* ap = agg + (size_t)n * S;
  if (S == 16) {
    float4* xv = (float4*)xp;
    const float4* av = (const float4*)ap;
#pragma unroll
    for (int i = 0; i < 4; ++i) {
      float4 xx = xv[i], aa = av[i];
      xx.x = ca * xx.x + cb * aa.x;
      xx.y = ca * xx.y + cb * aa.y;
      xx.z = ca * xx.z + cb * aa.z;
      xx.w = ca * xx.w + cb * aa.w;
      xv[i] = xx;
    }
  } else {
    for (int s = 0; s < S; ++s) xp[s] = ca * xp[s] + cb * ap[s];
  }
}

// ---------------------------------------------------------------------------
// Kernel 5: out[s][n] = softplus(post_diag[n]) * x[n][s] + mean[n]
// ---------------------------------------------------------------------------
__global__ void vd_final(const float* __restrict__ x, const float* __restrict__ pdiag,
                         const float* __restrict__ meanp, float* __restrict__ out,
                         int n_nodes, int S) {
  int n = blockIdx.x * blockDim.x + threadIdx.x;
  if (n >= n_nodes) return;
  float sp = vd_softplus(pdiag[n]);
  float m = meanp[n];
  const float* xp = x + (size_t)n * S;
  for (int s = 0; s < S; ++s)
    out[(size_t)s * n_nodes + n] = sp * xp[s] + m;
}

// ---------------------------------------------------------------------------
extern "C" void kernel_launch(void* const* d_in, const int* in_sizes, int n_in,
                              void* d_out, int out_size, void* d_ws, size_t ws_size,
                              hipStream_t stream) {
  const float* sample = (const float*)d_in[0];   // [S, N]
  const float* diag   = (const float*)d_in[1];   // [N]
  const float* pdiag  = (const float*)d_in[2];   // [N]
  const float* meanp  = (const float*)d_in[3];   // [N]
  const float* a1     = (const float*)d_in[4];   // [L]
  const float* a2     = (const float*)d_in[5];   // [L]
  const float* gp     = (const float*)d_in[6];   // [L]
  const int*   ei     = (const int*)d_in[7];     // [2, E] int32

  const int n_nodes   = in_sizes[1];
  const int S         = in_sizes[0] / n_nodes;
  const long long E   = (long long)in_sizes[7] / 2;
  const int L         = in_sizes[4];

  // workspace layout (all 16B aligned)
  char* ws = (char*)d_ws;
  int2*  pairs = (int2*)ws;
  size_t off = (size_t)E * sizeof(int2);
  off = (off + 15) & ~(size_t)15;
  float* x = (float*)(ws + off);
  off += (size_t)n_nodes * S * sizeof(float);
  off = (off + 15) & ~(size_t)15;
  float* agg = (float*)(ws + off);
  off += (size_t)n_nodes * S * sizeof(float);
  off = (off + 15) & ~(size_t)15;
  float* deg = (float*)(ws + off);

  const int B = VD_B;  // 8 wave32 waves per block
  const int nodeBlocks = (n_nodes + B - 1) / B;
  const long long edgeBlocks = (E + B - 1) / B;
  const long long edgeChunkBlocks = (E + (long long)B * VD_ILP - 1) / ((long long)B * VD_ILP);

  (void)hipMemsetAsync(deg, 0, (size_t)n_nodes * sizeof(float), stream);
  vd_prep_edges<<<(unsigned)edgeBlocks, B, 0, stream>>>(ei, pairs, deg, E);
  vd_init_x<<<nodeBlocks, B, 0, stream>>>(sample, diag, x, n_nodes, S);

  for (int l = 0; l < L; ++l) {
    (void)hipMemsetAsync(agg, 0, (size_t)n_nodes * S * sizeof(float), stream);
    if (S == 16)
      vd_scatter16<<<(unsigned)edgeChunkBlocks, B, 0, stream>>>(pairs, x, agg, E);
    else
      vd_scatter_gen<<<(unsigned)edgeBlocks, B, 0, stream>>>(pairs, x, agg, E, S);
    vd_combine<<<nodeBlocks, B, 0, stream>>>(x, agg, deg, a1, a2, gp, l, n_nodes, S);
  }

  vd_final<<<nodeBlocks, B, 0, stream>>>(x, pdiag, meanp, (float*)d_out, n_nodes, S);
}